// FactorizedSpectralConv_13469017440452
// MI455X (gfx1250) — compile-verified
//
#include <hip/hip_runtime.h>

// FNO FactorizedSpectralConv for MI455X (gfx1250, wave32).
// Pipeline: k_temb (WMMA f32 dense) -> k_fft_fwd (packed real FFT, LDS radix-2)
//        -> k_mix (per-mode complex GEMM via V_WMMA_F32_16X16X4_F32)
//        -> k_ifft (sparse-spectrum inverse packed real FFT).

#define B_    16
#define CIN   64
#define COUT  64
#define N_    16384
#define MODES 2048
#define TEMB  512
#define N2    8192      // N_/2, complex FFT length
#define LOGN2 13

#define PI_F 3.14159265358979323846f

typedef __attribute__((ext_vector_type(2))) float v2f;
typedef __attribute__((ext_vector_type(8))) float v8f;

static __device__ __forceinline__ float2 cmul(float c, float s, float2 v) {
    // (c + i s) * v
    return make_float2(c * v.x - s * v.y, c * v.y + s * v.x);
}

static __device__ __forceinline__ float silu_fast(float x) {
    // x * sigmoid(x) with native rcp (v_rcp_f32) instead of IEEE divide.
    return x * __builtin_amdgcn_rcpf(1.0f + __expf(-x));
}

// ---------------------------------------------------------------------------
// t[b, cin] = silu(temb[b, :]) @ dense_w[cin, :]^T + dense_b[cin]
// One workgroup, 4 waves; wave w owns output-column tile [16w, 16w+16).
// A = silu(temb): 16x512 fp32, B = dense_w^T: 512x64. K-loop in steps of 4
// with V_WMMA_F32_16X16X4_F32.
// ---------------------------------------------------------------------------
__global__ __launch_bounds__(128) void k_temb(const float* __restrict__ temb,
                                              const float* __restrict__ dw,
                                              const float* __restrict__ db,
                                              float* __restrict__ tout) {
    const int lane = threadIdx.x & 31;
    const int wave = threadIdx.x >> 5;   // 0..3
    const int o0   = wave * 16;
    const int hi   = lane >> 4;          // lane half: selects K {0,1} vs {2,3}
    const int mr   = lane & 15;          // A row (batch)
    const int nl   = lane & 15;          // B/D column

    v8f acc = {};
    #pragma unroll 4
    for (int k0 = 0; k0 < TEMB; k0 += 4) {
        const int kk = k0 + 2 * hi;
        // A operand: silu(temb[mr][kk]), silu(temb[mr][kk+1])
        float2 tv = *(const float2*)(temb + mr * TEMB + kk);
        v2f a;
        a.x = silu_fast(tv.x);
        a.y = silu_fast(tv.y);
        // B operand: dense_w[o0+nl][kk], dense_w[o0+nl][kk+1]
        float2 wv = *(const float2*)(dw + (o0 + nl) * TEMB + kk);
        v2f b;
        b.x = wv.x;
        b.y = wv.y;
        acc = __builtin_amdgcn_wmma_f32_16x16x4_f32(false, a, false, b,
                                                    (short)0, acc, false, false);
    }
    const float bias = db[o0 + nl];
    #pragma unroll
    for (int r = 0; r < 8; ++r) {
        const int M = r + 8 * hi;                    // batch row
        tout[M * CIN + o0 + nl] = acc[r] + bias;
    }
}

// ---------------------------------------------------------------------------
// Forward: one workgroup per (b, cin) row.
// Pack x[2t], x[2t+1] as complex, 8192-pt in-place DIT FFT in LDS,
// unpack modes 0..2047 of the 16384-pt rfft, add t[b,cin] to real part.
// Output layout mode-major: xm[(m*16 + b)*64 + cin]  (separate re/im planes).
// ---------------------------------------------------------------------------
__global__ __launch_bounds__(256) void k_fft_fwd(const float* __restrict__ x,
                                                 const float* __restrict__ tbuf,
                                                 float* __restrict__ xr,
                                                 float* __restrict__ xi) {
    __shared__ float2 Z[N2];   // 64 KB
    const int row = blockIdx.x;          // b*CIN + c
    const int b   = row >> 6;
    const int c   = row & 63;
    const int tid = threadIdx.x;

    // Load packed complex with bit-reversed addressing (coalesced float2 reads).
    const float2* xin = (const float2*)(x + (size_t)row * N_);
    for (int t = tid; t < N2; t += 256) {
        const int r = (int)(__brev((unsigned)t) >> (32 - LOGN2));
        Z[r] = xin[t];
    }
    __syncthreads();

    // 13 radix-2 DIT stages, forward twiddles e^{-i pi pos/half}.
    for (int s = 0; s < LOGN2; ++s) {
        const int   half = 1 << s;
        const float wstp = -PI_F / (float)half;
        for (int i = tid; i < N2 / 2; i += 256) {
            const int pos = i & (half - 1);
            const int i0  = ((i >> s) << (s + 1)) + pos;
            float ss, cc;
            __sincosf(wstp * (float)pos, &ss, &cc);
            float2 u  = Z[i0];
            float2 v  = Z[i0 + half];
            float2 wv = cmul(cc, ss, v);
            Z[i0]        = make_float2(u.x + wv.x, u.y + wv.y);
            Z[i0 + half] = make_float2(u.x - wv.x, u.y - wv.y);
        }
        __syncthreads();
    }

    // Real-FFT unpack for k = 0..MODES-1:
    //   Xe = (Z[k] + conj(Z[N2-k]))/2 ; Xo = (Z[k] - conj(Z[N2-k]))/(2i)
    //   X[k] = Xe + e^{-i*pi*k/N2} * Xo
    const float tval = tbuf[b * CIN + c];
    for (int m = tid; m < MODES; m += 256) {
        float2 Z1 = Z[m];
        float2 Z2 = Z[(N2 - m) & (N2 - 1)];
        Z2.y = -Z2.y;                                     // conj
        float2 Xe = make_float2(0.5f * (Z1.x + Z2.x), 0.5f * (Z1.y + Z2.y));
        float2 T  = make_float2(0.5f * (Z1.x - Z2.x), 0.5f * (Z1.y - Z2.y)); // = i*Xo
        float2 Xo = make_float2(T.y, -T.x);
        float ss, cc;
        __sincosf(-PI_F * (float)m / (float)N2, &ss, &cc);
        float2 E  = cmul(cc, ss, Xo);
        const size_t idx = ((size_t)m * B_ + b) * CIN + c;
        xr[idx] = Xe.x + E.x + tval;   // temb injection on the real part
        xi[idx] = Xe.y + E.y;
    }
}

// ---------------------------------------------------------------------------
// Spectral mixing: one workgroup per mode m (4 waves, wave -> 16-wide o tile).
// Complex GEMM [16 x 64] * [64 x 64] with fp32 WMMA, 4 wmma per K-step:
//   Dr += Ar*Br ; Dr += (-Ai)*Bi (VALU-negated Ai; fp32 WMMA has no neg_a)
//   Di += Ar*Bi ; Di += Ai*Br
// Weights read in native [i][o][m] layout (wr+wi = 67 MB, resident in 192MB L2);
// next K-step's lane-strided weight lines are prefetched (global_prefetch_b8)
// so their L2 latency hides behind the current step's 4 WMMAs.
// ---------------------------------------------------------------------------
__global__ __launch_bounds__(128) void k_mix(const float* __restrict__ xr,
                                             const float* __restrict__ xi,
                                             const float* __restrict__ wr,
                                             const float* __restrict__ wi,
                                             float* __restrict__ outr,
                                             float* __restrict__ outi) {
    const int m    = blockIdx.x;
    const int lane = threadIdx.x & 31;
    const int wave = threadIdx.x >> 5;
    const int o0   = wave * 16;
    const int hi   = lane >> 4;
    const int mr   = lane & 15;
    const int nl   = lane & 15;

    const float* Ar = xr + (size_t)m * B_ * CIN;   // [16][64] contiguous
    const float* Ai = xi + (size_t)m * B_ * CIN;

    v8f dr = {}, di = {};
    #pragma unroll
    for (int k0 = 0; k0 < CIN; k0 += 4) {
        const int kk = k0 + 2 * hi;
        float2 arv = *(const float2*)(Ar + mr * CIN + kk);
        float2 aiv = *(const float2*)(Ai + mr * CIN + kk);
        v2f a_r;  a_r.x  =  arv.x; a_r.y  =  arv.y;
        v2f a_i;  a_i.x  =  aiv.x; a_i.y  =  aiv.y;
        v2f a_in; a_in.x = -aiv.x; a_in.y = -aiv.y;   // -Ai (fp32 WMMA lacks neg_a)

        const size_t w0 = ((size_t)kk * COUT + (o0 + nl)) * MODES + m;
        const size_t w1 = ((size_t)(kk + 1) * COUT + (o0 + nl)) * MODES + m;
        v2f b_r; b_r.x = wr[w0]; b_r.y = wr[w1];
        v2f b_i; b_i.x = wi[w0]; b_i.y = wi[w1];

        // Prefetch next K-step's weight lines (L2-resident, lane-strided 8KB).
        if (k0 + 4 < CIN) {
            const size_t p0 = ((size_t)(kk + 4) * COUT + (o0 + nl)) * MODES + m;
            __builtin_prefetch(wr + p0, 0, 1);
            __builtin_prefetch(wi + p0, 0, 1);
        }

        dr = __builtin_amdgcn_wmma_f32_16x16x4_f32(false, a_r,  false, b_r,
                                                   (short)0, dr, false, false);
        dr = __builtin_amdgcn_wmma_f32_16x16x4_f32(false, a_in, false, b_i,
                                                   (short)0, dr, false, false);
        di = __builtin_amdgcn_wmma_f32_16x16x4_f32(false, a_r,  false, b_i,
                                                   (short)0, di, false, false);
        di = __builtin_amdgcn_wmma_f32_16x16x4_f32(false, a_i,  false, b_r,
                                                   (short)0, di, false, false);
    }

    #pragma unroll
    for (int r = 0; r < 8; ++r) {
        const int M = r + 8 * hi;                       // batch row
        const size_t idx = ((size_t)m * B_ + M) * COUT + o0 + nl;
        outr[idx] = dr[r];
        outi[idx] = di[r];
    }
}

// ---------------------------------------------------------------------------
// Inverse: one workgroup per (b, cout) row.
// Rebuild packed spectrum Z[k] from the 2048 nonzero rfft modes:
//   Xe = (X[k] + conj(X[N2-k]))/2 ; Xo = e^{+i*pi*k/N2}*(X[k]-conj(X[N2-k]))/2
//   Z[k] = Xe + i*Xo
// Inverse 8192-pt DIT (conjugate twiddles), scale 1/N2, unpack even/odd.
// ---------------------------------------------------------------------------
__global__ __launch_bounds__(256) void k_ifft(const float* __restrict__ outr,
                                              const float* __restrict__ outi,
                                              float* __restrict__ y) {
    __shared__ float2 Z[N2];   // 64 KB
    const int row = blockIdx.x;          // b*COUT + o
    const int b   = row >> 6;
    const int o   = row & 63;
    const int tid = threadIdx.x;

    for (int k = tid; k < N2; k += 256) {
        float2 Xk = make_float2(0.f, 0.f);
        float2 Xc = make_float2(0.f, 0.f);
        if (k < MODES) {
            const size_t idx = ((size_t)k * B_ + b) * COUT + o;
            Xk = make_float2(outr[idx], outi[idx]);
        }
        const int kc = N2 - k;                    // k==0 -> 8192 (zero mode)
        if (kc < MODES) {
            const size_t idx = ((size_t)kc * B_ + b) * COUT + o;
            Xc = make_float2(outr[idx], -outi[idx]);   // conj(X[N2-k])
        }
        float2 Xe = make_float2(0.5f * (Xk.x + Xc.x), 0.5f * (Xk.y + Xc.y));
        float2 Bv = make_float2(0.5f * (Xk.x - Xc.x), 0.5f * (Xk.y - Xc.y));
        float ss, cc;
        __sincosf(PI_F * (float)k / (float)N2, &ss, &cc);
        float2 Xo = cmul(cc, ss, Bv);
        float2 Zk = make_float2(Xe.x - Xo.y, Xe.y + Xo.x);  // Xe + i*Xo
        const int r = (int)(__brev((unsigned)k) >> (32 - LOGN2));
        Z[r] = Zk;
    }
    __syncthreads();

    // Inverse DIT: twiddles e^{+i pi pos/half}.
    for (int s = 0; s < LOGN2; ++s) {
        const int   half = 1 << s;
        const float wstp = PI_F / (float)half;
        for (int i = tid; i < N2 / 2; i += 256) {
            const int pos = i & (half - 1);
            const int i0  = ((i >> s) << (s + 1)) + pos;
            float ss, cc;
            __sincosf(wstp * (float)pos, &ss, &cc);
            float2 u  = Z[i0];
            float2 v  = Z[i0 + half];
            float2 wv = cmul(cc, ss, v);
            Z[i0]        = make_float2(u.x + wv.x, u.y + wv.y);
            Z[i0 + half] = make_float2(u.x - wv.x, u.y - wv.y);
        }
        __syncthreads();
    }

    // Unpack: x[2t] = Re z[t]/N2, x[2t+1] = Im z[t]/N2 (coalesced float2 stores).
    float2* yout = (float2*)(y + (size_t)row * N_);
    const float sc = 1.0f / (float)N2;
    for (int t = tid; t < N2; t += 256) {
        float2 z = Z[t];
        yout[t] = make_float2(z.x * sc, z.y * sc);
    }
}

// ---------------------------------------------------------------------------
extern "C" void kernel_launch(void* const* d_in, const int* in_sizes, int n_in,
                              void* d_out, int out_size, void* d_ws, size_t ws_size,
                              hipStream_t stream) {
    const float* x    = (const float*)d_in[0];  // [B, CIN, N]
    const float* temb = (const float*)d_in[1];  // [B, TEMB]
    const float* wr   = (const float*)d_in[2];  // [CIN, COUT, MODES]
    const float* wi   = (const float*)d_in[3];  // [CIN, COUT, MODES]
    const float* dw   = (const float*)d_in[4];  // [CIN, TEMB]
    const float* db   = (const float*)d_in[5];  // [CIN]
    float* out = (float*)d_out;                 // [B, COUT, N]

    char*  ws  = (char*)d_ws;
    size_t off = 0;
    float* tbuf = (float*)(ws + off); off += 4096;                       // 16*64 f32
    float* xr   = (float*)(ws + off); off += (size_t)MODES * B_ * CIN * 4;   // 8 MB
    float* xi   = (float*)(ws + off); off += (size_t)MODES * B_ * CIN * 4;   // 8 MB
    float* orr  = (float*)(ws + off); off += (size_t)MODES * B_ * COUT * 4;  // 8 MB
    float* oii  = (float*)(ws + off);                                        // 8 MB

    k_temb   <<<1,          128, 0, stream>>>(temb, dw, db, tbuf);
    k_fft_fwd<<<B_ * CIN,   256, 0, stream>>>(x, tbuf, xr, xi);
    k_mix    <<<MODES,      128, 0, stream>>>(xr, xi, wr, wi, orr, oii);
    k_ifft   <<<B_ * COUT,  256, 0, stream>>>(orr, oii, out);
}